// ResBlockGnn_64080912056839
// MI455X (gfx1250) — compile-verified
//
#include <hip/hip_runtime.h>

// GCN residual chain on MI455X (gfx1250, wave32).
// GEMMs use V_WMMA_F32_16X16X4_F32 (native fp32 matrix op); edge scatter is a
// wave-per-edge L2-atomic pattern (hW and acc both fit in the 192MB L2).

#define H       128
#define NCLS    16
#define NGRAPH  512

typedef __attribute__((ext_vector_type(2))) float v2f;
typedef __attribute__((ext_vector_type(8))) float v8f;

__global__ __launch_bounds__(256) void k_fill(float* __restrict__ p, float v, int n) {
  int i = blockIdx.x * 256 + threadIdx.x;
  if (i < n) p[i] = v;
}

// deg[dst] += 1 per edge (deg pre-filled with 1.0 for the self loop)
__global__ __launch_bounds__(256) void k_deg(const int* __restrict__ ei, float* __restrict__ deg, int E) {
  int e = blockIdx.x * 256 + threadIdx.x;
  if (e < E) atomicAdd(&deg[ei[E + e]], 1.0f);
}

// deg -> d^{-1/2} in place
__global__ __launch_bounds__(256) void k_dinv(float* __restrict__ deg, int n) {
  int i = blockIdx.x * 256 + threadIdx.x;
  if (i < n) { float d = deg[i]; deg[i] = d > 0.0f ? rsqrtf(d) : 0.0f; }
}

// Out[M,128] = (A1 [+ A2]) @ W  via V_WMMA_F32_16X16X4_F32.
// One wave per 16x16 tile; block of 8 waves covers all 128 columns for a 16-row stripe.
// A 16x4 layout: lanes 0-15 rows M, v[0]=K0,v[1]=K1; lanes 16-31 v[0]=K2,v[1]=K3.
// B 4x16 mirrored; C/D: VGPR r -> M = r + 8*(lane>=16), N = lane&15.
// HAS_A2 is a compile-time template parameter so the residual add is branchless
// (no per-iteration exec-mask save/restore around the WMMA).
template <bool HAS_A2>
__global__ __launch_bounds__(256) void k_gemm_wmma(const float* __restrict__ A1,
                                                   const float* __restrict__ A2,
                                                   const float* __restrict__ W,
                                                   float* __restrict__ Out) {
  const int lane = threadIdx.x & 31;
  const int wave = threadIdx.x >> 5;
  const int half = lane >> 4;        // 0: K pair {0,1}, 1: K pair {2,3}
  const int l16  = lane & 15;
  const int row0 = blockIdx.x << 4;
  const int col  = (wave << 4) + l16;

  const float* a1 = A1 + (size_t)(row0 + l16) * H;
  const float* a2 = HAS_A2 ? (A2 + (size_t)(row0 + l16) * H) : nullptr;

  v8f acc = {};
#pragma unroll 4
  for (int k0 = 0; k0 < H; k0 += 4) {
    const int ka = k0 + 2 * half;
    v2f a = *(const v2f*)(a1 + ka);            // global_load_b64
    if (HAS_A2) {
      v2f r = *(const v2f*)(a2 + ka);
      a[0] += r[0];                            // v_pk_add_f32, branchless
      a[1] += r[1];
    }
    v2f b;
    b[0] = W[(size_t)ka * H + col];
    b[1] = W[(size_t)(ka + 1) * H + col];
    acc = __builtin_amdgcn_wmma_f32_16x16x4_f32(false, a, false, b,
                                                (short)0, acc, false, false);
  }
#pragma unroll
  for (int r = 0; r < 8; ++r)
    Out[(size_t)(row0 + (half << 3) + r) * H + col] = acc[r];
}

// acc = bias + self-loop contribution (hW[i] * dinv[i]^2)
__global__ __launch_bounds__(256) void k_init_acc(float* __restrict__ acc,
                                                  const float* __restrict__ hW,
                                                  const float* __restrict__ bias,
                                                  const float* __restrict__ dinv, int nh) {
  int i = blockIdx.x * 256 + threadIdx.x;
  if (i >= nh) return;
  int node = i >> 7, f = i & (H - 1);
  float di = dinv[node];
  acc[i] = bias[f] + hW[i] * di * di;
}

// One wave per edge: src/dst/norm wave-uniform, each lane moves a float4 (512B/wave).
__global__ __launch_bounds__(256) void k_scatter(const int* __restrict__ ei,
                                                 const float* __restrict__ hW,
                                                 const float* __restrict__ dinv,
                                                 float* __restrict__ acc, int E) {
  int wid  = (int)((blockIdx.x * 256 + threadIdx.x) >> 5);
  int lane = threadIdx.x & 31;
  if (wid >= E) return;
  int src = ei[wid];
  int dst = ei[E + wid];
  float nrm = dinv[src] * dinv[dst];
  const float4* s = (const float4*)(hW + (size_t)src * H);
  float4 v = s[lane];
  float* d = acc + (size_t)dst * H + lane * 4;
  atomicAdd(d + 0, v.x * nrm);   // non-returning -> global_atomic_add_f32
  atomicAdd(d + 1, v.y * nrm);
  atomicAdd(d + 2, v.z * nrm);
  atomicAdd(d + 3, v.w * nrm);
}

__global__ __launch_bounds__(256) void k_relu(float* __restrict__ p, int n) {
  int i = blockIdx.x * 256 + threadIdx.x;
  if (i < n) p[i] = fmaxf(p[i], 0.0f);
}

__global__ __launch_bounds__(256) void k_count(const int* __restrict__ batch,
                                               float* __restrict__ counts, int n) {
  int i = blockIdx.x * 256 + threadIdx.x;
  if (i < n) atomicAdd(&counts[batch[i]], 1.0f);
}

__global__ __launch_bounds__(256) void k_poolsum(const int* __restrict__ batch,
                                                 const float* __restrict__ xc,
                                                 float* __restrict__ sums, int nh) {
  int i = blockIdx.x * 256 + threadIdx.x;
  if (i >= nh) return;
  int node = i >> 7, f = i & (H - 1);
  atomicAdd(&sums[(size_t)batch[node] * H + f], xc[i]);
}

__global__ __launch_bounds__(256) void k_mean(const float* __restrict__ sums,
                                              const float* __restrict__ counts,
                                              float* __restrict__ meanOut, int gh) {
  int i = blockIdx.x * 256 + threadIdx.x;
  if (i >= gh) return;
  int g = i >> 7;
  meanOut[i] = sums[i] / fmaxf(counts[g], 1.0f);
}

// y[g,c] = mean[g] . W_lin[:,c] + b_lin[c]   (512x16, trivial)
__global__ __launch_bounds__(256) void k_linear(const float* __restrict__ mean,
                                                const float* __restrict__ Wl,
                                                const float* __restrict__ bl,
                                                float* __restrict__ y, int gc) {
  int i = blockIdx.x * 256 + threadIdx.x;
  if (i >= gc) return;
  int g = i / NCLS, c = i - g * NCLS;
  const float* m = mean + (size_t)g * H;
  float s = bl[c];
#pragma unroll 8
  for (int k = 0; k < H; ++k) s += m[k] * Wl[k * NCLS + c];
  y[i] = s;
}

extern "C" void kernel_launch(void* const* d_in, const int* in_sizes, int n_in,
                              void* d_out, int out_size, void* d_ws, size_t ws_size,
                              hipStream_t stream) {
  const float* x     = (const float*)d_in[0];
  const int*   ei    = (const int*)d_in[1];
  const int*   batch = (const int*)d_in[2];
  const float* W_in  = (const float*)d_in[3];
  const float* b_in  = (const float*)d_in[4];
  const float* W_hid = (const float*)d_in[5];
  const float* b_hid = (const float*)d_in[6];
  const float* W_lin = (const float*)d_in[7];
  const float* b_lin = (const float*)d_in[8];

  const int N  = in_sizes[0] / H;        // 50000 (multiple of 16)
  const int E  = in_sizes[1] / 2;        // 800000
  const int NL = in_sizes[5] / (H * H);  // 4
  const int NH = N * H;

  // workspace layout (floats)
  float* ws   = (float*)d_ws;
  size_t off  = 0;
  float* dinv = ws + off; off += (size_t)N; off = (off + 63) & ~(size_t)63;
  float* B0   = ws + off; off += (size_t)NH;
  float* B1   = ws + off; off += (size_t)NH;
  float* B2   = ws + off; off += (size_t)NH;
  float* HW   = ws + off; off += (size_t)NH;
  float* counts = ws + off; off += NGRAPH;       // counts then sums contiguous
  float* sums   = ws + off; off += (size_t)NGRAPH * H;

  const dim3 b256(256);
  auto cdiv = [](int a, int b) { return (a + b - 1) / b; };

  // ---- normalization: deg = 1 (self loop) + indeg ; dinv = rsqrt(deg) ----
  k_fill<<<cdiv(N, 256), b256, 0, stream>>>(dinv, 1.0f, N);
  k_deg <<<cdiv(E, 256), b256, 0, stream>>>(ei, dinv, E);
  k_dinv<<<cdiv(N, 256), b256, 0, stream>>>(dinv, N);

  // ---- 5 GCN layers (input + 4 residual hidden), rotating buffers ----
  float* cur = nullptr; float* pre = nullptr; float* freeb = nullptr;
  for (int layer = 0; layer <= NL; ++layer) {
    const float* Wl = (layer == 0) ? W_in : W_hid + (size_t)(layer - 1) * H * H;
    const float* bl = (layer == 0) ? b_in : b_hid + (size_t)(layer - 1) * H;
    float* acc      = (layer == 0) ? B0 : freeb;

    if (layer == 0 || pre == nullptr) {
      const float* A1 = (layer == 0) ? x : cur;
      k_gemm_wmma<false><<<N / 16, b256, 0, stream>>>(A1, nullptr, Wl, HW);
    } else {
      k_gemm_wmma<true><<<N / 16, b256, 0, stream>>>(cur, pre, Wl, HW);
    }
    k_init_acc <<<cdiv(NH, 256), b256, 0, stream>>>(acc, HW, bl, dinv, NH);
    k_scatter  <<<cdiv(E, 8),    b256, 0, stream>>>(ei, HW, dinv, acc, E);
    k_relu     <<<cdiv(NH, 256), b256, 0, stream>>>(acc, NH);

    if (layer == 0) { cur = B0; pre = nullptr; freeb = B1; }
    else { float* oldpre = pre; pre = cur; cur = acc; freeb = oldpre ? oldpre : B2; }
  }

  // ---- global mean pool + final linear ----
  k_fill   <<<cdiv(NGRAPH * (H + 1), 256), b256, 0, stream>>>(counts, 0.0f, NGRAPH * (H + 1));
  k_count  <<<cdiv(N, 256),  b256, 0, stream>>>(batch, counts, N);
  k_poolsum<<<cdiv(NH, 256), b256, 0, stream>>>(batch, cur, sums, NH);

  float* yOut    = (float*)d_out;                 // [512,16]
  float* meanOut = (float*)d_out + NGRAPH * NCLS; // [512,128]
  k_mean  <<<cdiv(NGRAPH * H, 256),    b256, 0, stream>>>(sums, counts, meanOut, NGRAPH * H);
  k_linear<<<cdiv(NGRAPH * NCLS, 256), b256, 0, stream>>>(meanOut, W_lin, b_lin, yOut, NGRAPH * NCLS);
}